// KalmanFilter_61160334295739
// MI455X (gfx1250) — compile-verified
//
#include <hip/hip_runtime.h>
#include <math.h>

typedef float v2f __attribute__((ext_vector_type(2)));
typedef float v8f __attribute__((ext_vector_type(8)));

#define BATCHN 128
#define TLEN   256
#define DZ     32
#define DA     16
#define DU     4
#define KK     8
#define GHD    64
#define DOBS   16
#define DIN    192

// ---------------- WMMA f32 16x16x4 helpers (CDNA5 VGPR layouts, ISA 7.12.2) -----------

__device__ __forceinline__ v8f wmma_f32x4(v2f a, v2f b, v8f c) {
#if __has_builtin(__builtin_amdgcn_wmma_f32_16x16x4_f32)
  // 8 args: (neg_a, A, neg_b, B, c_mod, C, reuse_a, reuse_b)
  return __builtin_amdgcn_wmma_f32_16x16x4_f32(false, a, false, b, (short)0, c, false, false);
#else
  return c; // host pass / unsupported target: never executed on device
#endif
}

// A fragment: rows m0..m0+15, cols k0..k0+3 of row-major src (ld).
// Layout: lanes 0-15 -> (M=lane, K=k0 / k0+1); lanes 16-31 -> (M=lane-16, K=k0+2 / k0+3)
__device__ __forceinline__ v2f ldfrag_a(const float* src, int ld, int m0, int k0, int lane) {
  int m  = m0 + (lane & 15);
  int kb = k0 + ((lane >> 4) << 1);
  const float* p = src + m * ld + kb;
  v2f a; a.x = p[0]; a.y = p[1];
  return a;
}
// B fragment: rows k0..k0+3, cols n0..n0+15 of row-major src (ld).
__device__ __forceinline__ v2f ldfrag_b(const float* src, int ld, int k0, int n0, int lane) {
  int n  = n0 + (lane & 15);
  int kb = k0 + ((lane >> 4) << 1);
  v2f b; b.x = src[kb * ld + n]; b.y = src[(kb + 1) * ld + n];
  return b;
}
// B = srcT^T (i.e. logical B[k][n] = srcT[n][k]); srcT row-major (N x K) with ld.
__device__ __forceinline__ v2f ldfrag_bT(const float* srcT, int ld, int k0, int n0, int lane) {
  int n  = n0 + (lane & 15);
  int kb = k0 + ((lane >> 4) << 1);
  const float* p = srcT + n * ld + kb;
  v2f b; b.x = p[0]; b.y = p[1];
  return b;
}
// C/D tile store: VGPR r -> (M = m0 + r + 8*(lane>=16), N = n0 + lane%16)
__device__ __forceinline__ void st_tile(float* D, int ldd, int m0, int n0, int lane, v8f c) {
  int n  = n0 + (lane & 15);
  int mb = m0 + ((lane >> 4) << 3);
#pragma unroll
  for (int r = 0; r < 8; ++r) D[(mb + r) * ldd + n] = c[r];
}

__device__ __forceinline__ v8f gemm_nn(const float* A, int lda, int m0,
                                       const float* B, int ldb, int n0,
                                       int K, int lane, v8f acc) {
#pragma unroll
  for (int k0 = 0; k0 < K; k0 += 4)
    acc = wmma_f32x4(ldfrag_a(A, lda, m0, k0, lane), ldfrag_b(B, ldb, k0, n0, lane), acc);
  return acc;
}
__device__ __forceinline__ v8f gemm_nt(const float* A, int lda, int m0,
                                       const float* Bt, int ldbt, int n0,
                                       int K, int lane, v8f acc) {
#pragma unroll
  for (int k0 = 0; k0 < K; k0 += 4)
    acc = wmma_f32x4(ldfrag_a(A, lda, m0, k0, lane), ldfrag_bT(Bt, ldbt, k0, n0, lane), acc);
  return acc;
}

__device__ __forceinline__ float sigmoidf_(float x) { return 1.f / (1.f + __expf(-x)); }

// -------------------------------- main scan kernel ------------------------------------
// One wave32 per batch row; grid = 128 blocks x 32 threads. All per-row state in LDS.

__global__ __launch_bounds__(32) void kf_scan_kernel(
    const float* __restrict__ a_seq, const float* __restrict__ h_obs,
    const float* __restrict__ Amat,  const float* __restrict__ Cmat,
    const float* __restrict__ Bmat,  const float* __restrict__ u_seq,
    const float* __restrict__ mask,  const float* __restrict__ P0,
    const float* __restrict__ matQ,  const float* __restrict__ matR,
    const float* __restrict__ Wx,    const float* __restrict__ Wh,
    const float* __restrict__ gb,    const float* __restrict__ outW,
    const float* __restrict__ outb,  float* __restrict__ out)
{
  const int lane = threadIdx.x;
  const int b    = blockIdx.x;

  __shared__ float sP[DZ*DZ], sPf[DZ*DZ], sT[DZ*DZ], sAk[DZ*DZ], sIKC[DZ*DZ];
  __shared__ float sL[DZ*DZ], sRaw[DZ*DZ], sQ[DZ*DZ];
  __shared__ float sCk[DA*DZ], sCp[DA*DZ], sKg[DZ*DA], sT1[DA*DZ];
  __shared__ float sAug[DA*48], sS[DA*DA], sR[DA*DA];
  __shared__ float sZj[KK*DZ], sAj[KK*DA], sXin[DIN];
  __shared__ float sZ[DZ], sZn[DZ], sZloc[DZ], sZpred[DZ], sGh[GHD];
  __shared__ float sAlpha[KK], sAimm[KK], sLl[KK], sRk[DA], sAt[DA], sU[DU], sBk[DZ*DU];

  const size_t BT = (size_t)BATCHN * TLEN;
  float* o_zf    = out;
  float* o_Pf    = o_zf    + BT * DZ;
  float* o_zloc  = o_Pf    + BT * DZ * DZ;
  float* o_tril  = o_zloc  + BT * DZ;
  float* o_zpred = o_tril  + BT * DZ * DZ;
  float* o_Pp    = o_zpred + BT * DZ;
  float* o_af    = o_Pp    + BT * DZ * DZ;
  float* o_ap    = o_af    + BT * DA;
  float* o_S     = o_ap    + BT * DA;
  float* o_alpha = o_S     + BT * DA * DA;
  float* o_aimm  = o_alpha + BT * KK;
  float* o_R     = o_aimm  + BT * KK;
  float* o_Q     = o_R     + DA * DA;

  const v8f z8 = {0.f, 0.f, 0.f, 0.f, 0.f, 0.f, 0.f, 0.f};

  // ---- one-time per block: Q = mQ mQ^T + 1e-3 I, R = mR mR^T + 1e-3 I, init state ----
  for (int it = 0; it < 32; ++it) {
    int idx = lane + 32 * it; int i = idx >> 5, j = idx & 31;
    float s = 0.f;
    for (int k = 0; k < DZ; ++k) s += matQ[i*DZ + k] * matQ[j*DZ + k];
    sQ[idx] = s + (i == j ? 1e-3f : 0.f);
  }
  for (int it = 0; it < 8; ++it) {
    int idx = lane + 32 * it; int i = idx >> 4, j = idx & 15;
    float s = 0.f;
    for (int k = 0; k < DA; ++k) s += matR[i*DA + k] * matR[j*DA + k];
    sR[idx] = s + (i == j ? 1e-3f : 0.f);
  }
  for (int it = 0; it < 32; ++it) sP[lane + 32*it]  = P0[lane + 32*it];
  for (int it = 0; it < 16; ++it) sCp[lane + 32*it] = Cmat[lane + 32*it]; // C_matrices[0]
  sZ[lane] = 0.f;
  sGh[lane] = 0.f; sGh[lane + 32] = 0.f;
  __syncthreads();
  if (b == 0) {
    for (int it = 0; it < 8;  ++it) o_R[lane + 32*it] = sR[lane + 32*it];
    for (int it = 0; it < 32; ++it) o_Q[lane + 32*it] = sQ[lane + 32*it];
  }

  // ------------------------------------ time scan -------------------------------------
  for (int t = 0; t < TLEN; ++t) {
    const size_t bt = (size_t)b * TLEN + t;
    const float mk = mask[bt];
    if (lane < DA) sAt[lane] = a_seq[bt * DA + lane];
    if (lane < DU) sU[lane]  = u_seq[bt * DU + lane];
    __syncthreads();

    // zj[k,i] = A[k] @ z_prev
    for (int it = 0; it < 8; ++it) {
      int idx = lane + 32 * it; int k = idx >> 5, i = idx & 31;
      float s = 0.f;
      for (int j = 0; j < DZ; ++j) s += Amat[k*DZ*DZ + i*DZ + j] * sZ[j];
      sZj[idx] = s;
    }
    __syncthreads();
    // aj[k,a] = C[k] @ zj[k]
    for (int it = 0; it < 4; ++it) {
      int idx = lane + 32 * it; int k = idx >> 4, a = idx & 15;
      float s = 0.f;
      for (int i = 0; i < DZ; ++i) s += Cmat[k*DA*DZ + a*DZ + i] * sZj[k*DZ + i];
      sAj[idx] = s;
    }
    __syncthreads();
    // log_lik + alpha_imm = softmax(log_lik) * mk
    if (lane < KK) {
      float s = 0.f;
      for (int a = 0; a < DA; ++a) { float d = sAt[a] - sAj[lane*DA + a]; s += d * d; }
      sLl[lane] = -s;
    }
    __syncthreads();
    if (lane < KK) {
      float m = sLl[0];
      for (int k = 1; k < KK; ++k) m = fmaxf(m, sLl[k]);
      float den = 0.f;
      for (int k = 0; k < KK; ++k) den += __expf(sLl[k] - m);
      sAimm[lane] = __expf(sLl[lane] - m) / den * mk;
    }
    // x_in = [a_prev(16) | h_obs(16) | z_prev(32) | aj(128)]
    if (lane < DA) {
      float s = 0.f;
      for (int i = 0; i < DZ; ++i) s += sCp[lane*DZ + i] * sZ[i];
      sXin[lane] = s;
      sXin[DA + lane] = h_obs[(size_t)b * DOBS + lane];
    }
    sXin[32 + lane] = sZ[lane];
    for (int it = 0; it < 4; ++it) sXin[64 + lane + 32*it] = sAj[lane + 32*it];
    __syncthreads();

    // ---- GRU cell: each lane owns columns j = lane + 32*it ----
    float gx[6], gl[6];
#pragma unroll
    for (int it = 0; it < 6; ++it) {
      int j = lane + 32 * it;
      float s = gb[j];
      for (int i = 0; i < DIN; ++i) s += sXin[i] * Wx[i*DIN + j];
      gx[it] = s;
      float s2 = 0.f;
      for (int i = 0; i < GHD; ++i) s2 += sGh[i] * Wh[i*DIN + j];
      gl[it] = s2;
    }
    float hnew[2];
#pragma unroll
    for (int it = 0; it < 2; ++it) {
      int j = lane + 32 * it;
      float r  = sigmoidf_(gx[it]     + gl[it]);
      float zg = sigmoidf_(gx[it + 2] + gl[it + 2]);
      float n  = tanhf(gx[it + 4] + r * gl[it + 4]);
      hnew[it] = (1.f - zg) * n + zg * sGh[j];
    }
    __syncthreads();
    sGh[lane] = hnew[0]; sGh[lane + 32] = hnew[1];
    __syncthreads();

    // alpha = softmax(gh @ out_W + out_b)
    if (lane < KK) {
      float s = outb[lane];
      for (int j = 0; j < GHD; ++j) s += sGh[j] * outW[j*KK + lane];
      sLl[lane] = s;
    }
    __syncthreads();
    if (lane < KK) {
      float m = sLl[0];
      for (int k = 1; k < KK; ++k) m = fmaxf(m, sLl[k]);
      float den = 0.f;
      for (int k = 0; k < KK; ++k) den += __expf(sLl[k] - m);
      sAlpha[lane] = __expf(sLl[lane] - m) / den;
    }
    __syncthreads();

    // ---- mixture matrices A_k, C_k, B_k ----
    float al[KK];
#pragma unroll
    for (int k = 0; k < KK; ++k) al[k] = sAlpha[k];
    for (int it = 0; it < 32; ++it) {
      int idx = lane + 32 * it; float s = 0.f;
#pragma unroll
      for (int k = 0; k < KK; ++k) s += al[k] * Amat[k*DZ*DZ + idx];
      sAk[idx] = s;
    }
    for (int it = 0; it < 16; ++it) {
      int idx = lane + 32 * it; float s = 0.f;
#pragma unroll
      for (int k = 0; k < KK; ++k) s += al[k] * Cmat[k*DA*DZ + idx];
      sCk[idx] = s;
    }
    for (int it = 0; it < 4; ++it) {
      int idx = lane + 32 * it; float s = 0.f;
#pragma unroll
      for (int k = 0; k < KK; ++k) s += al[k] * Bmat[k*DZ*DU + idx];
      sBk[idx] = s;
    }
    __syncthreads();

    // z = A_k z_prev ; r_k = a_t - C_k z
    {
      float s = 0.f;
      for (int j = 0; j < DZ; ++j) s += sAk[lane*DZ + j] * sZ[j];
      sZn[lane] = s;
    }
    __syncthreads();
    if (lane < DA) {
      float s = 0.f;
      for (int i = 0; i < DZ; ++i) s += sCk[lane*DZ + i] * sZn[i];
      sRk[lane] = sAt[lane] - s;
    }
    __syncthreads();

    // ---- WMMA: T1 = Ck@P ; S = T1@Ck^T ; M2 = Ck@P^T -> aug[:,16:48] ----
    for (int n0 = 0; n0 < DZ; n0 += 16)
      st_tile(sT1, DZ, 0, n0, lane, gemm_nn(sCk, DZ, 0, sP, DZ, n0, DZ, lane, z8));
    __syncthreads();
    st_tile(sS, DA, 0, 0, lane, gemm_nt(sT1, DZ, 0, sCk, DZ, 0, DZ, lane, z8));
    for (int n0 = 0; n0 < DZ; n0 += 16)
      st_tile(sAug, 48, 0, 16 + n0, lane, gemm_nt(sCk, DZ, 0, sP, DZ, n0, DZ, lane, z8));
    __syncthreads();
    // S += R + 1e-4 I
    for (int it = 0; it < 8; ++it) {
      int idx = lane + 32 * it; int i = idx >> 4, j = idx & 15;
      sS[idx] += sR[idx] + (i == j ? 1e-4f : 0.f);
    }
    __syncthreads();
    // aug[:, :16] = S^T
    for (int it = 0; it < 8; ++it) {
      int idx = lane + 32 * it; int r = idx >> 4, c = idx & 15;
      sAug[r*48 + c] = sS[c*DA + r];
    }
    __syncthreads();

    // ---- Gauss-Jordan: solve S^T X = (Ck P^T)  (16 x 48 augmented) ----
    for (int p = 0; p < DA; ++p) {
      float ipiv = 1.f / sAug[p*48 + p];
      __syncthreads();
      sAug[p*48 + lane] *= ipiv;
      if (lane < 16) sAug[p*48 + 32 + lane] *= ipiv;
      __syncthreads();
      float f = (lane < DA && lane != p) ? sAug[lane*48 + p] : 0.f;
      __syncthreads();
      if (lane < DA && lane != p)
        for (int c = 0; c < 48; ++c) sAug[lane*48 + c] -= f * sAug[p*48 + c];
      __syncthreads();
    }
    // Kg = X^T * mk   (32x16, ld 16)
    for (int it = 0; it < 16; ++it) {
      int idx = lane + 32 * it; int i = idx >> 4, a = idx & 15;
      sKg[idx] = sAug[a*48 + 16 + i] * mk;
    }
    __syncthreads();

    // ---- IKC = I - Kg @ Ck (WMMA) ----
    for (int m0 = 0; m0 < DZ; m0 += 16)
      for (int n0 = 0; n0 < DZ; n0 += 16)
        st_tile(sIKC, DZ, m0, n0, lane, gemm_nn(sKg, DA, m0, sCk, DZ, n0, DA, lane, z8));
    __syncthreads();
    for (int it = 0; it < 32; ++it) {
      int idx = lane + 32 * it; int i = idx >> 5, j = idx & 31;
      sIKC[idx] = (i == j ? 1.f : 0.f) - sIKC[idx];
    }
    __syncthreads();

    // ---- P_f = sym(IKC P IKC^T + Kg R Kg^T) + 1e-3 I (WMMA, chained accumulators) ----
    for (int m0 = 0; m0 < DZ; m0 += 16)
      for (int n0 = 0; n0 < DZ; n0 += 16)
        st_tile(sT, DZ, m0, n0, lane, gemm_nn(sIKC, DZ, m0, sP, DZ, n0, DZ, lane, z8));
    for (int m0 = 0; m0 < DZ; m0 += 16)
      st_tile(sT1, DA, m0, 0, lane, gemm_nn(sKg, DA, m0, sR, DA, 0, DA, lane, z8)); // T3 = Kg@R
    __syncthreads();
    for (int m0 = 0; m0 < DZ; m0 += 16)
      for (int n0 = 0; n0 < DZ; n0 += 16) {
        v8f acc = gemm_nt(sT,  DZ, m0, sIKC, DZ, n0, DZ, lane, z8);
        acc     = gemm_nt(sT1, DA, m0, sKg,  DA, n0, DA, lane, acc);
        st_tile(sRaw, DZ, m0, n0, lane, acc);
      }
    __syncthreads();
    for (int it = 0; it < 32; ++it) {
      int idx = lane + 32 * it; int i = idx >> 5, j = idx & 31;
      sPf[idx] = 0.5f * (sRaw[idx] + sRaw[j*DZ + i]) + (i == j ? 1e-3f : 0.f);
    }
    __syncthreads();

    // z_loc = z + Kg r_k
    {
      float s = sZn[lane];
      for (int a = 0; a < DA; ++a) s += sKg[lane*DA + a] * sRk[a];
      sZloc[lane] = s;
    }
    __syncthreads();

    // ---- Cholesky(P_f), right-looking, lane-per-row ----
    for (int it = 0; it < 32; ++it) sL[lane + 32*it] = sPf[lane + 32*it];
    __syncthreads();
    for (int j = 0; j < DZ; ++j) {
      float d   = sqrtf(sL[j*DZ + j]);
      float inv = 1.f / d;
      float lij = (lane > j) ? sL[lane*DZ + j] * inv : 0.f;
      __syncthreads();
      if (lane == j) sL[j*DZ + j] = d;
      if (lane >  j) sL[lane*DZ + j] = lij;
      __syncthreads();
      if (lane > j)
        for (int c = j + 1; c <= lane; ++c) sL[lane*DZ + c] -= lij * sL[c*DZ + j];
      __syncthreads();
    }
    for (int it = 0; it < 32; ++it) {
      int idx = lane + 32 * it; int i = idx >> 5, j = idx & 31;
      if (j > i) sL[idx] = 0.f;
    }
    __syncthreads();

    // z_pred = A_k z_filt + B_k u
    {
      float s = 0.f;
      for (int j = 0; j < DZ; ++j) s += sAk[lane*DZ + j] * sZloc[j];
      for (int u = 0; u < DU; ++u) s += sBk[lane*DU + u] * sU[u];
      sZpred[lane] = s;
    }
    __syncthreads();

    // ---- P_p = sym(A_k P_f A_k^T) + Q + 1e-3 I (WMMA) -> new state P ----
    for (int m0 = 0; m0 < DZ; m0 += 16)
      for (int n0 = 0; n0 < DZ; n0 += 16)
        st_tile(sT, DZ, m0, n0, lane, gemm_nn(sAk, DZ, m0, sPf, DZ, n0, DZ, lane, z8));
    __syncthreads();
    for (int m0 = 0; m0 < DZ; m0 += 16)
      for (int n0 = 0; n0 < DZ; n0 += 16)
        st_tile(sRaw, DZ, m0, n0, lane, gemm_nt(sT, DZ, m0, sAk, DZ, n0, DZ, lane, z8));
    __syncthreads();
    for (int it = 0; it < 32; ++it) {
      int idx = lane + 32 * it; int i = idx >> 5, j = idx & 31;
      sP[idx] = 0.5f * (sRaw[idx] + sRaw[j*DZ + i]) + sQ[idx] + (i == j ? 1e-3f : 0.f);
    }
    __syncthreads();

    // ---- outputs ----
    o_zf   [bt*DZ + lane] = sZloc[lane];
    o_zloc [bt*DZ + lane] = sZloc[lane];
    o_zpred[bt*DZ + lane] = sZpred[lane];
    for (int it = 0; it < 32; ++it) {
      int idx = lane + 32 * it;
      o_Pf  [bt*DZ*DZ + idx] = sPf[idx];
      o_tril[bt*DZ*DZ + idx] = sL[idx];
      o_Pp  [bt*DZ*DZ + idx] = sP[idx];
    }
    if (lane < DA) {
      float sf = 0.f, sp = 0.f;
      for (int i = 0; i < DZ; ++i) {
        sf += sCk[lane*DZ + i] * sZloc[i];
        sp += sCk[lane*DZ + i] * sZpred[i];
      }
      o_af[bt*DA + lane] = sf;
      o_ap[bt*DA + lane] = sp;
    }
    for (int it = 0; it < 8; ++it) o_S[bt*DA*DA + lane + 32*it] = sS[lane + 32*it];
    if (lane < KK) { o_alpha[bt*KK + lane] = sAlpha[lane]; o_aimm[bt*KK + lane] = sAimm[lane]; }

    // ---- carry ----
    sZ[lane] = sZloc[lane];
    for (int it = 0; it < 16; ++it) sCp[lane + 32*it] = sCk[lane + 32*it];
    __syncthreads();
  }
}

extern "C" void kernel_launch(void* const* d_in, const int* in_sizes, int n_in,
                              void* d_out, int out_size, void* d_ws, size_t ws_size,
                              hipStream_t stream) {
  (void)in_sizes; (void)n_in; (void)out_size; (void)d_ws; (void)ws_size;
  const float* a_seq = (const float*)d_in[0];
  const float* h_obs = (const float*)d_in[1];
  const float* Amat  = (const float*)d_in[2];
  const float* Cmat  = (const float*)d_in[3];
  const float* Bmat  = (const float*)d_in[4];
  const float* u_seq = (const float*)d_in[5];
  const float* maskp = (const float*)d_in[6];
  const float* P0    = (const float*)d_in[7];
  const float* matQ  = (const float*)d_in[8];
  const float* matR  = (const float*)d_in[9];
  const float* Wx    = (const float*)d_in[10];
  const float* Wh    = (const float*)d_in[11];
  const float* gbv   = (const float*)d_in[12];
  const float* outW  = (const float*)d_in[13];
  const float* outb  = (const float*)d_in[14];
  kf_scan_kernel<<<dim3(BATCHN), dim3(32), 0, stream>>>(
      a_seq, h_obs, Amat, Cmat, Bmat, u_seq, maskp, P0, matQ, matR,
      Wx, Wh, gbv, outW, outb, (float*)d_out);
}